// MA_MSE_32762010534318
// MI455X (gfx1250) — compile-verified
//
#include <hip/hip_runtime.h>
#include <hip/hip_bf16.h>

#define BB   32
#define SS   8
#define FEA  2560
#define TT   800
#define NCL  512
#define EMB  256
#define HH   8

typedef __attribute__((ext_vector_type(2))) float v2f;
typedef __attribute__((ext_vector_type(8))) float v8f;

// ---------------------------------------------------------------------------
// Kernel 1: masked pool over time.  pooled[b,s,f] = sum_t x[b,f,t]*mask[b,s,t]
// Bandwidth-bound stage (262 MB of x). Mask for batch b staged in LDS once
// per block (25.6 KB), reused by all 8 feature rows handled by the block.
// One wave per feature row; lanes stride t by 128 with float4 loads.
// ---------------------------------------------------------------------------
__global__ __launch_bounds__(256) void pool_kernel(const float* __restrict__ x,
                                                   const float* __restrict__ mask,
                                                   float* __restrict__ pooled) {
    __shared__ float mlds[SS * TT];                    // 6400 floats = 25.6 KB
    const int b     = blockIdx.x / (FEA / 8);
    const int fbase = (blockIdx.x % (FEA / 8)) * 8;

    for (int i = threadIdx.x; i < SS * TT; i += 256)
        mlds[i] = mask[(size_t)b * SS * TT + i];
    __syncthreads();

    const int wave = threadIdx.x >> 5;
    const int lane = threadIdx.x & 31;
    const int f    = fbase + wave;
    const float* __restrict__ xrow = x + ((size_t)b * FEA + f) * TT;

    float acc[SS] = {};
    for (int t0 = lane * 4; t0 < TT; t0 += 128) {      // TT % 4 == 0, covers [0,800)
        float4 xv = *(const float4*)(xrow + t0);
#pragma unroll
        for (int s = 0; s < SS; ++s) {
            float4 mv = *(const float4*)(&mlds[s * TT + t0]);
            acc[s] += xv.x * mv.x + xv.y * mv.y + xv.z * mv.z + xv.w * mv.w;
        }
    }
#pragma unroll
    for (int s = 0; s < SS; ++s) {
        float v = acc[s];
#pragma unroll
        for (int off = 16; off >= 1; off >>= 1) v += __shfl_xor(v, off, 32);
        if (lane == 0) pooled[((size_t)b * SS + s) * FEA + f] = v;
    }
}

// ---------------------------------------------------------------------------
// Kernel 2: per (b,s): denom = sum_t mask; w[bs,h] = (pooled/denom) . W_w[h] + W_b[h]
// One block per (b,s); block-wide reductions in LDS.
// ---------------------------------------------------------------------------
__global__ __launch_bounds__(256) void wscore_kernel(const float* __restrict__ pooled,
                                                     const float* __restrict__ mask,
                                                     const float* __restrict__ W_w,
                                                     const float* __restrict__ W_b,
                                                     float* __restrict__ wout) {
    const int bs = blockIdx.x;
    __shared__ float red[256];

    float d = 0.f;
    for (int t = threadIdx.x; t < TT; t += 256)
        d += mask[(size_t)bs * TT + t];
    red[threadIdx.x] = d;
    __syncthreads();
    for (int k = 128; k >= 1; k >>= 1) {
        if (threadIdx.x < k) red[threadIdx.x] += red[threadIdx.x + k];
        __syncthreads();
    }
    const float inv = 1.0f / (red[0] + 1e-10f);
    __syncthreads();

    float acc[HH] = {};
    for (int f = threadIdx.x; f < FEA; f += 256) {
        float xv = pooled[(size_t)bs * FEA + f] * inv;
#pragma unroll
        for (int h = 0; h < HH; ++h) acc[h] += xv * W_w[h * FEA + f];
    }
#pragma unroll
    for (int h = 0; h < HH; ++h) {
        red[threadIdx.x] = acc[h];
        __syncthreads();
        for (int k = 128; k >= 1; k >>= 1) {
            if (threadIdx.x < k) red[threadIdx.x] += red[threadIdx.x + k];
            __syncthreads();
        }
        if (threadIdx.x == 0) wout[bs * HH + h] = red[0] + W_b[h];
        __syncthreads();
    }
}

// ---------------------------------------------------------------------------
// Kernel 3: u[n,h] = m[n,:] . U_w[h,:] + U_b[h]   (512x8, trivially small)
// ---------------------------------------------------------------------------
__global__ __launch_bounds__(256) void u_kernel(const float* __restrict__ m,
                                                const float* __restrict__ U_w,
                                                const float* __restrict__ U_b,
                                                float* __restrict__ u) {
    const int idx = blockIdx.x * 256 + threadIdx.x;    // 0..4095
    const int n = idx / HH, h = idx % HH;
    float acc = 0.f;
    for (int e = 0; e < EMB; ++e) acc += m[n * EMB + e] * U_w[h * EMB + e];
    u[idx] = acc + U_b[h];
}

// ---------------------------------------------------------------------------
// Kernel 4: a[bs,n] = sigmoid(sum_h v[h]*tanh(w[bs,h]+u[n,h]) + v_b)
// ---------------------------------------------------------------------------
__global__ __launch_bounds__(256) void gate_kernel(const float* __restrict__ w,
                                                   const float* __restrict__ u,
                                                   const float* __restrict__ v_w,
                                                   const float* __restrict__ v_b,
                                                   float* __restrict__ a) {
    const int bs = blockIdx.x;
    __shared__ float wl[HH], vl[HH], vb;
    if (threadIdx.x < HH) {
        wl[threadIdx.x] = w[bs * HH + threadIdx.x];
        vl[threadIdx.x] = v_w[threadIdx.x];
    }
    if (threadIdx.x == 0) vb = v_b[0];
    __syncthreads();

    for (int n = threadIdx.x; n < NCL; n += 256) {
        float c = vb;
#pragma unroll
        for (int h = 0; h < HH; ++h) c += vl[h] * tanhf(wl[h] + u[n * HH + h]);
        a[(size_t)bs * NCL + n] = 1.0f / (1.0f + __expf(-c));
    }
}

// ---------------------------------------------------------------------------
// Kernel 5: e = a @ m  -> (256 x 512) x (512 x 256), fp32 via
// V_WMMA_F32_16X16X4_F32. One wave per 16x16 output tile, K-loop of 128 WMMAs.
// A 16x4 layout: lane L holds M=L%16, vec elem j = K = 2*(L/16)+j.
// B 4x16 layout (symmetric): lane L holds N=L%16, vec elem j = K = 2*(L/16)+j.
// C/D: VGPR r of lane L = [M = r + 8*(L/16), N = L%16].
// ---------------------------------------------------------------------------
__global__ __launch_bounds__(256) void egemm_kernel(const float* __restrict__ A,
                                                    const float* __restrict__ Bm,
                                                    float* __restrict__ E) {
    const int wave = threadIdx.x >> 5;
    const int lane = threadIdx.x & 31;
    const int tile = blockIdx.x * 8 + wave;            // 0..255
    const int mi = tile >> 4, ni = tile & 15;
    const int half = lane >> 4;                        // 0 or 1
    const int lm   = lane & 15;

    const int arow = mi * 16 + lm;
    const int bcol = ni * 16 + lm;

    v8f c = {};
    for (int k0 = 0; k0 < NCL; k0 += 4) {
        const int ka = k0 + 2 * half;
        v2f av;
        av.x = A[(size_t)arow * NCL + ka];
        av.y = A[(size_t)arow * NCL + ka + 1];
        v2f bv;
        bv.x = Bm[(size_t)ka * EMB + bcol];
        bv.y = Bm[(size_t)(ka + 1) * EMB + bcol];
        c = __builtin_amdgcn_wmma_f32_16x16x4_f32(false, av, false, bv,
                                                  (short)0, c, false, false);
    }
#pragma unroll
    for (int r = 0; r < 8; ++r) {
        const int row = mi * 16 + r + 8 * half;
        E[(size_t)row * EMB + bcol] = c[r];
    }
}

// ---------------------------------------------------------------------------
extern "C" void kernel_launch(void* const* d_in, const int* in_sizes, int n_in,
                              void* d_out, int out_size, void* d_ws, size_t ws_size,
                              hipStream_t stream) {
    const float* x    = (const float*)d_in[0];   // (B,FEA,T)
    const float* mask = (const float*)d_in[1];   // (B,S,T)
    const float* W_w  = (const float*)d_in[2];   // (H,FEA)
    const float* W_b  = (const float*)d_in[3];   // (H,)
    const float* U_w  = (const float*)d_in[4];   // (H,EMB)
    const float* U_b  = (const float*)d_in[5];   // (H,)
    const float* v_w  = (const float*)d_in[6];   // (1,H)
    const float* v_b  = (const float*)d_in[7];   // (1,)
    const float* m    = (const float*)d_in[8];   // (N,EMB)
    float* e          = (float*)d_out;           // (B,S,EMB)

    float* wsF    = (float*)d_ws;
    float* pooled = wsF;                         // B*S*FEA   = 655360
    float* wbuf   = pooled + (size_t)BB * SS * FEA;   // 2048
    float* ubuf   = wbuf + BB * SS * HH;              // 4096
    float* abuf   = ubuf + NCL * HH;                  // 131072

    pool_kernel  <<<BB * (FEA / 8), 256, 0, stream>>>(x, mask, pooled);
    wscore_kernel<<<BB * SS,        256, 0, stream>>>(pooled, mask, W_w, W_b, wbuf);
    u_kernel     <<<(NCL * HH) / 256, 256, 0, stream>>>(m, U_w, U_b, ubuf);
    gate_kernel  <<<BB * SS,        256, 0, stream>>>(wbuf, ubuf, v_w, v_b, abuf);
    egemm_kernel <<<(BB * SS * (EMB / 16)) / 128, 256, 0, stream>>>(abuf, m, e);
}